// JaggedArgmaxModule_30150670418630
// MI455X (gfx1250) — compile-verified
//
#include <hip/hip_runtime.h>
#include <hip/hip_bf16.h>
#include <stdint.h>

#define WAVES_PER_BLOCK 8
#define MAX_STAGE_CHUNKS 4           // 4 chunks * 128 floats = 512 floats per wave
#define CHUNK_ELEMS 128              // 32 lanes * 4 floats (one b128 per lane)

// Jagged argmax: one wave32 per segment.
// Fast path stages the segment into LDS with gfx1250 async global->LDS b128
// copies (ASYNCcnt), then reduces out of LDS. Fallback path (unaligned start
// or len > 639) uses direct strided global loads. First-occurrence tie-break
// is enforced with an explicit (value, position) comparison everywhere.
__global__ __launch_bounds__(256) void jagged_argmax_kernel(
    const float* __restrict__ values,
    const long long* __restrict__ prefix,   // int64 inclusive ends
    int n_seg,
    int* __restrict__ out)
{
    __shared__ float stage[WAVES_PER_BLOCK][MAX_STAGE_CHUNKS * CHUNK_ELEMS];

    const int lane = threadIdx.x & 31;
    const int wv   = threadIdx.x >> 5;
    const int seg  = blockIdx.x * WAVES_PER_BLOCK + wv;
    if (seg >= n_seg) return;

    const long long start = (seg == 0) ? 0LL : prefix[seg - 1];
    const long long end   = prefix[seg];
    const int len = (int)(end - start);
    if (len <= 0) {
        if (lane == 0) out[seg] = 0;
        return;
    }
    const float* base = values + start;

    float bestV = -__builtin_inff();
    int   bestP = 0x7fffffff;

    int nFull = len >> 7;  // number of full 128-element chunks
    const bool fast = ((start & 3) == 0) && (nFull <= MAX_STAGE_CHUNKS);
    if (!fast) nFull = 0;  // everything goes through the fallback loop

    if (nFull > 0) {
        // LDS byte offset of this lane's 16B slot inside this wave's region.
        // Flat-LDS rule: LDS offset == low 32 bits of the generic address.
        float* dst0 = &stage[wv][lane * 4];
        const unsigned ldsOff = (unsigned)(uintptr_t)dst0;
        const float* src0 = base + lane * 4;

        // Fire all chunk copies (one async b128 per lane per chunk), then one
        // wait. ASYNCcnt increments once per instruction per wave.
        for (int c = 0; c < nFull; ++c) {
            const unsigned loff = ldsOff + (unsigned)(c * (CHUNK_ELEMS * 4));
            const float*   gsrc = src0 + (size_t)c * CHUNK_ELEMS;
            asm volatile("global_load_async_to_lds_b128 %0, %1, off"
                         :
                         : "v"(loff), "v"(gsrc)
                         : "memory");
        }
        asm volatile("s_wait_asynccnt 0x0" ::: "memory");

        // Read back (ds_load_b128) and scan. Positions are segment-relative.
        for (int c = 0; c < nFull; ++c) {
            const float4 v4 = *(const float4*)&stage[wv][c * CHUNK_ELEMS + lane * 4];
            const int p0 = c * CHUNK_ELEMS + lane * 4;
            const float e[4] = {v4.x, v4.y, v4.z, v4.w};
#pragma unroll
            for (int j = 0; j < 4; ++j) {
                const float v = e[j];
                const int   p = p0 + j;
                if (v > bestV || (v == bestV && p < bestP)) { bestV = v; bestP = p; }
            }
        }
    }

    // Tail (len % 128) and full fallback path: direct strided global loads.
    for (int p = (nFull << 7) + lane; p < len; p += 32) {
        const float v = base[p];
        if (v > bestV || (v == bestV && p < bestP)) { bestV = v; bestP = p; }
    }

    // Wave32 cross-lane (value, position) reduction, lower index wins ties.
#pragma unroll
    for (int off = 16; off > 0; off >>= 1) {
        const float ov = __shfl_xor(bestV, off, 32);
        const int   op = __shfl_xor(bestP, off, 32);
        if (ov > bestV || (ov == bestV && op < bestP)) { bestV = ov; bestP = op; }
    }

    if (lane == 0) out[seg] = bestP;
}

extern "C" void kernel_launch(void* const* d_in, const int* in_sizes, int n_in,
                              void* d_out, int out_size, void* d_ws, size_t ws_size,
                              hipStream_t stream) {
    const float*     values = (const float*)d_in[0];
    const long long* prefix = (const long long*)d_in[1];
    const int n_seg = in_sizes[1];

    int* out = (int*)d_out;  // reference output dtype is int32

    dim3 block(256);
    dim3 grid((n_seg + WAVES_PER_BLOCK - 1) / WAVES_PER_BLOCK);
    hipLaunchKernelGGL(jagged_argmax_kernel, grid, block, 0, stream,
                       values, prefix, n_seg, out);
}